// gen_color_stride_vgg16_77129022701766
// MI455X (gfx1250) — compile-verified
//
#include <hip/hip_runtime.h>
#include <hip/hip_bf16.h>

// ---------------------------------------------------------------------------
// Problem constants (from reference): B=4, C=64, H=W=224, S_t=S_r=257
// ---------------------------------------------------------------------------
#define BB 4
#define CC 64
#define HH 224
#define WW 224
#define SS 257          // segments (max label 256 inclusive)
#define SP 272          // padded to 17*16 for WMMA tiles
#define HW (HH * WW)
#define INTER (222 * 222)

typedef __attribute__((ext_vector_type(2))) float v2f;
typedef __attribute__((ext_vector_type(8))) float v8f;

// monotonic float <-> uint encoding so atomic max-on-uint == max-on-float
__device__ __forceinline__ unsigned enc_f(float f) {
    unsigned u = __float_as_uint(f);
    return (u & 0x80000000u) ? ~u : (u | 0x80000000u);
}
__device__ __forceinline__ float dec_f(unsigned e) {
    unsigned u = (e & 0x80000000u) ? (e ^ 0x80000000u) : ~e;
    return __uint_as_float(u);
}

// ---------------------------------------------------------------------------
// 1) Segment scatter-max over interior pixels. One block per (b,c,tensor).
//    LDS atomics (ds_max_u32), block owns its whole output slice.
// ---------------------------------------------------------------------------
__global__ void seg_pool_kernel(const float* __restrict__ f0, const float* __restrict__ f1,
                                const float* __restrict__ f2,
                                const int* __restrict__ lt, const int* __restrict__ lr,
                                unsigned* __restrict__ o0, unsigned* __restrict__ o1,
                                unsigned* __restrict__ o2) {
    const int which = blockIdx.y;                 // 0: target, 1: ref, 2: rgb_ref
    const float* feat = (which == 0) ? f0 : (which == 1) ? f1 : f2;
    const int* label = (which == 0) ? lt : lr;
    unsigned* out = (which == 0) ? o0 : (which == 1) ? o1 : o2;

    const int bc = blockIdx.x;                    // b*64 + c
    const int b = bc >> 6;

    __shared__ unsigned smax[SS];
    for (int s = threadIdx.x; s < SS; s += blockDim.x) smax[s] = 0u;
    __syncthreads();

    const float* fp = feat + (size_t)bc * HW;
    const int* lp = label + (size_t)b * HW;
    for (int p = threadIdx.x; p < INTER; p += blockDim.x) {
        int r = p / 222;
        int q = p - r * 222;
        int pix = (r + 1) * WW + (q + 1);
        float v = fp[pix];
        int s = lp[pix];
        atomicMax(&smax[s], enc_f(v));            // ds_max_u32
    }
    __syncthreads();

    unsigned* op = out + (size_t)bc * SS;
    for (int s = threadIdx.x; s < SS; s += blockDim.x) op[s] = smax[s];
}

// ---------------------------------------------------------------------------
// 2) Per-batch segment counts: n = max(label) + 1 (over FULL mask)
// ---------------------------------------------------------------------------
__global__ void label_max_kernel(const int* __restrict__ lt, const int* __restrict__ lr,
                                 int* __restrict__ nseg) {
    const int which = blockIdx.y;
    const int b = blockIdx.x;
    const int* lp = ((which == 0) ? lt : lr) + (size_t)b * HW;
    __shared__ int sm[256];
    int m = 0;
    for (int p = threadIdx.x; p < HW; p += 256) m = max(m, lp[p]);
    sm[threadIdx.x] = m;
    __syncthreads();
    for (int st = 128; st > 0; st >>= 1) {
        if (threadIdx.x < st) sm[threadIdx.x] = max(sm[threadIdx.x], sm[threadIdx.x + st]);
        __syncthreads();
    }
    if (threadIdx.x == 0) nseg[which * BB + b] = sm[0] + 1;
}

// ---------------------------------------------------------------------------
// 3) Decode encoded maxima -> padded f32 [B][64][272], empty segment -> 0
// ---------------------------------------------------------------------------
__global__ void decode_kernel(const unsigned* __restrict__ e0, const unsigned* __restrict__ e1,
                              const unsigned* __restrict__ e2,
                              float* __restrict__ x0, float* __restrict__ x1,
                              float* __restrict__ x2) {
    int i = blockIdx.x * blockDim.x + threadIdx.x;
    const int total = 3 * BB * CC * SP;
    if (i >= total) return;
    int which = i / (BB * CC * SP);
    int rem = i % (BB * CC * SP);
    int bc = rem / SP;
    int s = rem % SP;
    const unsigned* e = (which == 0) ? e0 : (which == 1) ? e1 : e2;
    float* x = (which == 0) ? x0 : (which == 1) ? x1 : x2;
    float v = 0.f;
    if (s < SS) {
        unsigned u = e[(size_t)bc * SS + s];
        if (u != 0u) v = dec_f(u);
    }
    x[(size_t)bc * SP + s] = v;
}

// ---------------------------------------------------------------------------
// 4) Channel norm (in place): per (b,s) subtract mean over C, divide by L2+eps
// ---------------------------------------------------------------------------
__global__ void chan_norm_kernel(float* __restrict__ x0, float* __restrict__ x1) {
    int i = blockIdx.x * blockDim.x + threadIdx.x;
    const int total = 2 * BB * SP;
    if (i >= total) return;
    int which = i / (BB * SP);
    int rem = i % (BB * SP);
    int b = rem / SP, s = rem % SP;
    float* x = ((which == 0) ? x0 : x1) + (size_t)b * CC * SP + s;
    float mean = 0.f;
    #pragma unroll 8
    for (int c = 0; c < CC; c++) mean += x[c * SP];
    mean *= (1.f / 64.f);
    float nrm = 0.f;
    #pragma unroll 8
    for (int c = 0; c < CC; c++) { float t = x[c * SP] - mean; nrm += t * t; }
    float inv = 1.f / (sqrtf(nrm) + 1e-5f);
    #pragma unroll 8
    for (int c = 0; c < CC; c++) x[c * SP] = (x[c * SP] - mean) * inv;
}

// ---------------------------------------------------------------------------
// 5) Projections q/k/v = W(64x64) @ X(64x272) + bias, WMMA f32 16x16x4.
//    One wave per 16x16 output tile; 16 K-steps.
//    A layout (16x4 f32): lane<16: v0=K0,v1=K1; lane>=16: v0=K2,v1=K3; M=lane&15
//    B layout (4x16):     v0: K=half*2, v1: K=half*2+1; N=lane&15
//    D layout (16x16):    vgpr i: M=i+half*8; N=lane&15
// ---------------------------------------------------------------------------
__global__ void proj_wmma_kernel(const float* __restrict__ xt, const float* __restrict__ xr,
                                 const float* __restrict__ xv,
                                 const float* __restrict__ Wq, const float* __restrict__ bq,
                                 const float* __restrict__ Wk, const float* __restrict__ bk,
                                 const float* __restrict__ Wv, const float* __restrict__ bv,
                                 float* __restrict__ qo, float* __restrict__ ko,
                                 float* __restrict__ vo) {
    const int s0 = blockIdx.x * 16;
    const int o0 = blockIdx.y * 16;
    const int b = blockIdx.z / 3;
    const int which = blockIdx.z % 3;
    const float* X = ((which == 0) ? xt : (which == 1) ? xr : xv) + (size_t)b * CC * SP;
    const float* W = (which == 0) ? Wq : (which == 1) ? Wk : Wv;
    const float* bias = (which == 0) ? bq : (which == 1) ? bk : bv;
    float* out = ((which == 0) ? qo : (which == 1) ? ko : vo) + (size_t)b * CC * SP;

    const int lane = threadIdx.x;
    const int half = lane >> 4;
    const int l15 = lane & 15;
    v8f acc = {};
    #pragma unroll
    for (int k0 = 0; k0 < CC; k0 += 4) {
        int ka = k0 + half * 2;
        v2f a, bmat;
        a.x = W[(o0 + l15) * CC + ka];
        a.y = W[(o0 + l15) * CC + ka + 1];
        bmat.x = X[(size_t)ka * SP + s0 + l15];
        bmat.y = X[(size_t)(ka + 1) * SP + s0 + l15];
        acc = __builtin_amdgcn_wmma_f32_16x16x4_f32(false, a, false, bmat, (short)0, acc,
                                                    false, false);
    }
    #pragma unroll
    for (int i = 0; i < 8; i++) {
        int row = o0 + i + half * 8;
        out[(size_t)row * SP + s0 + l15] = acc[i] + bias[row];
    }
}

// ---------------------------------------------------------------------------
// 6) sim[n][m] = (sum_c q[c][n]*k[c][m]) * 100, with ref-column mask -> -inf
// ---------------------------------------------------------------------------
__global__ void sim_wmma_kernel(const float* __restrict__ q, const float* __restrict__ k,
                                const int* __restrict__ nseg, float* __restrict__ simb) {
    const int n0 = blockIdx.x * 16;
    const int m0 = blockIdx.y * 16;
    const int b = blockIdx.z;
    const float* Q = q + (size_t)b * CC * SP;
    const float* K = k + (size_t)b * CC * SP;
    float* S = simb + (size_t)b * SP * SP;
    const int n_r = nseg[BB + b];

    const int lane = threadIdx.x;
    const int half = lane >> 4;
    const int l15 = lane & 15;
    v8f acc = {};
    #pragma unroll
    for (int c0 = 0; c0 < CC; c0 += 4) {
        int ca = c0 + half * 2;
        v2f a, bmat;
        a.x = Q[(size_t)ca * SP + n0 + l15];         // A[n][c] = q[c][n]
        a.y = Q[(size_t)(ca + 1) * SP + n0 + l15];
        bmat.x = K[(size_t)ca * SP + m0 + l15];      // B[c][m] = k[c][m]
        bmat.y = K[(size_t)(ca + 1) * SP + m0 + l15];
        acc = __builtin_amdgcn_wmma_f32_16x16x4_f32(false, a, false, bmat, (short)0, acc,
                                                    false, false);
    }
    #pragma unroll
    for (int i = 0; i < 8; i++) {
        int n = n0 + i + half * 8;
        int m = m0 + l15;
        float vv = acc[i] * 100.f;                    // temperature 0.01
        if (m >= SS || m >= n_r) vv = -__builtin_inff();
        S[(size_t)n * SP + m] = vv;
    }
}

// ---------------------------------------------------------------------------
// 7) Row softmax over m<257, then multiply row by tmask; zero padded rows/cols
// ---------------------------------------------------------------------------
__global__ void softmax_kernel(float* __restrict__ simb, const int* __restrict__ nseg) {
    const int n = blockIdx.x;   // 0..271
    const int b = blockIdx.y;
    float* row = simb + (size_t)b * SP * SP + (size_t)n * SP;
    const int tid = threadIdx.x;

    if (n >= SS) {              // padded rows -> zeros so wg WMMA can read them
        for (int m = tid; m < SP; m += 256) row[m] = 0.f;
        return;
    }
    __shared__ float red[256];
    const int n_t = nseg[b];

    float mx = -__builtin_inff();
    for (int m = tid; m < SS; m += 256) mx = fmaxf(mx, row[m]);
    red[tid] = mx;
    __syncthreads();
    for (int st = 128; st > 0; st >>= 1) {
        if (tid < st) red[tid] = fmaxf(red[tid], red[tid + st]);
        __syncthreads();
    }
    mx = red[0];
    __syncthreads();

    float sum = 0.f;
    for (int m = tid; m < SS; m += 256) {
        float e = __expf(row[m] - mx);  // exp(-inf)=0 for masked cols
        row[m] = e;
        sum += e;
    }
    red[tid] = sum;
    __syncthreads();
    for (int st = 128; st > 0; st >>= 1) {
        if (tid < st) red[tid] += red[tid + st];
        __syncthreads();
    }
    float scale = ((n < n_t) ? 1.f : 0.f) / red[0];
    for (int m = tid; m < SS; m += 256) row[m] *= scale;
    for (int m = SS + tid; m < SP; m += 256) row[m] = 0.f;   // padded cols
}

// ---------------------------------------------------------------------------
// 8) wg[c][n] = sum_m attn[n][m] * v[c][m]  (68 WMMA K-steps over padded m)
// ---------------------------------------------------------------------------
__global__ void wg_wmma_kernel(const float* __restrict__ attn, const float* __restrict__ v,
                               float* __restrict__ wg) {
    const int n0 = blockIdx.x * 16;
    const int c0 = blockIdx.y * 16;
    const int b = blockIdx.z;
    const float* A = attn + (size_t)b * SP * SP;
    const float* V = v + (size_t)b * CC * SP;
    float* W = wg + (size_t)b * CC * SP;

    const int lane = threadIdx.x;
    const int half = lane >> 4;
    const int l15 = lane & 15;
    v8f acc = {};
    #pragma unroll 4
    for (int m0 = 0; m0 < SP; m0 += 4) {
        int ma = m0 + half * 2;
        v2f a, bmat;
        a.x = A[(size_t)(n0 + l15) * SP + ma];        // A[n][m]
        a.y = A[(size_t)(n0 + l15) * SP + ma + 1];
        bmat.x = V[(size_t)(c0 + l15) * SP + ma];     // B[m][c] = v[c][m]
        bmat.y = V[(size_t)(c0 + l15) * SP + ma + 1];
        acc = __builtin_amdgcn_wmma_f32_16x16x4_f32(false, a, false, bmat, (short)0, acc,
                                                    false, false);
    }
    #pragma unroll
    for (int i = 0; i < 8; i++) {
        int n = n0 + i + half * 8;
        int c = c0 + l15;
        W[(size_t)c * SP + n] = acc[i];
    }
}

// ---------------------------------------------------------------------------
// 9) Unpool: out[b][c][h][w] = wg[b][c][label[b][h][w]]  (streaming 51MB write)
// ---------------------------------------------------------------------------
__global__ void unpool_kernel(const float* __restrict__ wg, const int* __restrict__ lt,
                              float* __restrict__ out) {
    long i = (long)blockIdx.x * blockDim.x + threadIdx.x;
    const long total = (long)BB * CC * HW;
    if (i >= total) return;
    int p = (int)(i % HW);
    long bc = i / HW;
    int b = (int)(bc >> 6);
    int s = lt[(size_t)b * HW + p];
    out[i] = wg[bc * SP + s];
}

// ---------------------------------------------------------------------------
// Launch
// ---------------------------------------------------------------------------
extern "C" void kernel_launch(void* const* d_in, const int* in_sizes, int n_in,
                              void* d_out, int out_size, void* d_ws, size_t ws_size,
                              hipStream_t stream) {
    const float* feat_t = (const float*)d_in[0];
    const float* feat_r = (const float*)d_in[1];
    const float* feat_v = (const float*)d_in[2];
    const float* Wq = (const float*)d_in[3];
    const float* bq = (const float*)d_in[4];
    const float* Wk = (const float*)d_in[5];
    const float* bk = (const float*)d_in[6];
    const float* Wv = (const float*)d_in[7];
    const float* bv = (const float*)d_in[8];
    const int* lt = (const int*)d_in[9];
    const int* lr = (const int*)d_in[10];
    float* out = (float*)d_out;

    // workspace carve-up (floats); total ~3.93 MB
    float* ws = (float*)d_ws;
    size_t off = 0;
    const size_t encN = (size_t)BB * CC * SS;   // 65792
    const size_t padN = (size_t)BB * CC * SP;   // 69632
    unsigned* enc_t = (unsigned*)(ws + off); off += encN;
    unsigned* enc_r = (unsigned*)(ws + off); off += encN;
    unsigned* enc_v = (unsigned*)(ws + off); off += encN;
    float* xt = ws + off; off += padN;          // normalized target feats
    float* xr = ws + off; off += padN;          // normalized ref feats
    float* xv = ws + off; off += padN;          // raw rgb-ref feats
    float* qb = ws + off; off += padN;
    float* kb = ws + off; off += padN;
    float* vb = ws + off; off += padN;
    float* simb = ws + off; off += (size_t)BB * SP * SP;   // 295936
    float* wgb = ws + off; off += padN;
    int* nseg = (int*)(ws + off); off += 8;     // n_t[4], n_r[4]

    // 1) segment scatter-max (3 tensors)
    seg_pool_kernel<<<dim3(BB * CC, 3), 256, 0, stream>>>(feat_t, feat_r, feat_v, lt, lr,
                                                          enc_t, enc_r, enc_v);
    // 2) per-batch counts
    label_max_kernel<<<dim3(BB, 2), 256, 0, stream>>>(lt, lr, nseg);
    // 3) decode to padded f32
    {
        int total = 3 * BB * CC * SP;
        decode_kernel<<<(total + 255) / 256, 256, 0, stream>>>(enc_t, enc_r, enc_v, xt, xr, xv);
    }
    // 4) channel norm (in place on xt, xr)
    {
        int total = 2 * BB * SP;
        chan_norm_kernel<<<(total + 255) / 256, 256, 0, stream>>>(xt, xr);
    }
    // 5) q/k/v projections (WMMA)
    proj_wmma_kernel<<<dim3(SP / 16, CC / 16, BB * 3), 32, 0, stream>>>(
        xt, xr, xv, Wq, bq, Wk, bk, Wv, bv, qb, kb, vb);
    // 6) sim (WMMA) + temperature + ref mask
    sim_wmma_kernel<<<dim3(SP / 16, SP / 16, BB), 32, 0, stream>>>(qb, kb, nseg, simb);
    // 7) softmax + target mask (attn stored back into simb)
    softmax_kernel<<<dim3(SP, BB), 256, 0, stream>>>(simb, nseg);
    // 8) wg = attn @ v^T (WMMA)
    wg_wmma_kernel<<<dim3(SP / 16, CC / 16, BB), 32, 0, stream>>>(simb, vb, wgb);
    // 9) unpool gather to pixel grid
    {
        long total = (long)BB * CC * HW;
        unpool_kernel<<<(unsigned)((total + 255) / 256), 256, 0, stream>>>(wgb, lt, out);
    }
    (void)in_sizes; (void)n_in; (void)out_size; (void)ws_size;
}